// GTQNetV0_68650757259460
// MI455X (gfx1250) — compile-verified
//
#include <hip/hip_runtime.h>
#include <hip/hip_bf16.h>

// Fused GTQNet forward for MI455X (gfx1250, wave32, WMMA).
// One block = one batch item (128 graph nodes). All activations stay in
// LDS/registers; weights (~110KB) are preformatted into f32 WMMA B-fragment
// layout in d_ws by tiny prep kernels, then stream from L2 every block.
// Math is exact f32 via V_WMMA_F32_16X16X4_F32 (reference is f32).

typedef float v2f __attribute__((ext_vector_type(2)));
typedef float v8f __attribute__((ext_vector_type(8)));

static __device__ __forceinline__ v8f wmma4(v2f a, v2f b, v8f c) {
  // D = A(16x4 f32) x B(4x16 f32) + C(16x16 f32)
  return __builtin_amdgcn_wmma_f32_16x16x4_f32(false, a, false, b, (short)0, c,
                                               false, false);
}

constexpr int LS = 66;   // LDS row stride in floats (even for b64, avoids 16-way bank conflicts)
constexpr int NTOK = 128;

// ---------------- weight preformat kernel ----------------
// B-fragment layout (per 16x4 step): element (K,N) -> lane = N%16 + 16*((K-K0)>>1),
// vgpr/component v = (K-K0)&1.  Stored as frag[nt*KSTEPS + k0i][lane*2 + v].
__global__ void prep_weights(const float* __restrict__ W, const float* __restrict__ bsrc,
                             float* __restrict__ outB, float* __restrict__ outBias,
                             int din, int dout, int dinp, int ntiles) {
  int ksteps = dinp >> 2;
  int Bsz = ntiles * ksteps * 64;
  int total = Bsz + ntiles * 16;
  int i = blockIdx.x * blockDim.x + threadIdx.x;
  if (i >= total) return;
  if (i < Bsz) {
    int frag = i >> 6;
    int e = i & 63;
    int lane = e >> 1;
    int v = e & 1;
    int nt = frag / ksteps;
    int k0 = (frag % ksteps) * 4;
    int n = nt * 16 + (lane & 15);
    int k = k0 + 2 * (lane >> 4) + v;
    outB[i] = (k < din && n < dout) ? W[k * dout + n] : 0.0f;
  } else {
    int j = i - Bsz;
    outBias[j] = (j < dout) ? bsrc[j] : 0.0f;
  }
}

// ---------------- device helpers ----------------
template <int NT>
static __device__ __forceinline__ void zeroAcc(v8f* a) {
#pragma unroll
  for (int nt = 0; nt < NT; ++nt) a[nt] = (v8f){};
}

// plain GEMM chunk: A rows from one LDS buffer
template <int KS, int NT>
static __device__ __forceinline__ void mm_plain(const float* aRow, const float* bBase, v8f* acc) {
#pragma unroll
  for (int k = 0; k < KS; ++k) {
    v2f a = *(const v2f*)(aRow + 4 * k);
#pragma unroll
    for (int nt = 0; nt < NT; ++nt) {
      v2f bb = *(const v2f*)(bBase + (nt * KS + k) * 64);
      acc[nt] = wmma4(a, bb, acc[nt]);
    }
  }
}

// two chains sharing the same A (stage-1 layer 1)
template <int KS, int NT>
static __device__ __forceinline__ void mm_dual(const float* aRow, const float* bR,
                                               const float* bS, v8f* accR, v8f* accS) {
#pragma unroll
  for (int k = 0; k < KS; ++k) {
    v2f a = *(const v2f*)(aRow + 4 * k);
#pragma unroll
    for (int nt = 0; nt < NT; ++nt) {
      v2f br = *(const v2f*)(bR + (nt * KS + k) * 64);
      accR[nt] = wmma4(a, br, accR[nt]);
      v2f bs = *(const v2f*)(bS + (nt * KS + k) * 64);
      accS[nt] = wmma4(a, bs, accS[nt]);
    }
  }
}

// combined A = root(own rows) + gate * sup(partner rows)  (the adj swap-add)
template <int KS, int NT>
static __device__ __forceinline__ void mm_comb(const float* aOwn, const float* aPart,
                                               float gate, const float* bBase, v8f* acc) {
#pragma unroll
  for (int k = 0; k < KS; ++k) {
    v2f a0 = *(const v2f*)(aOwn + 4 * k);
    v2f a1 = *(const v2f*)(aPart + 4 * k);
    v2f a = a0 + gate * a1;
#pragma unroll
    for (int nt = 0; nt < NT; ++nt) {
      v2f bb = *(const v2f*)(bBase + (nt * KS + k) * 64);
      acc[nt] = wmma4(a, bb, acc[nt]);
    }
  }
}

template <int NT, bool RELU>
static __device__ __forceinline__ void finish(v8f* acc, const float* biasBase, int n) {
#pragma unroll
  for (int nt = 0; nt < NT; ++nt) {
    float bb = biasBase[nt * 16 + n];
#pragma unroll
    for (int v = 0; v < 8; ++v) {
      float x = acc[nt][v] + bb;
      if (RELU) x = fmaxf(x, 0.0f);
      acc[nt][v] = x;
    }
  }
}

// D layout: lane holds (M = v + 8*(lane>>4), N = lane&15)
template <int NT>
static __device__ __forceinline__ void storeD(float* dstRow0, const v8f* acc, int lane) {
  int h = lane >> 4, n = lane & 15;
#pragma unroll
  for (int nt = 0; nt < NT; ++nt)
#pragma unroll
    for (int v = 0; v < 8; ++v)
      dstRow0[(v + 8 * h) * LS + nt * 16 + n] = acc[nt][v];
}

// ---------------- fused network kernel ----------------
__global__ __launch_bounds__(256) void gtq_fused(const float* __restrict__ s,
                                                 const float* __restrict__ g,
                                                 const float* __restrict__ wsf,
                                                 float* __restrict__ out) {
  extern __shared__ float sm[];
  float* buf0 = sm;                 // [128][LS]  root-chain activations
  float* buf1 = sm + NTOK * LS;     // [128][LS]  sup-chain activations
  float* sg = sm + 2 * NTOK * LS;   // [64] gates

  const int b = blockIdx.x;
  const int tid = threadIdx.x;
  const int lane = tid & 31;
  const int w = tid >> 5;           // wave id 0..7 -> token tile rows [16w,16w+16)

  // ---- build X = [pos0,pos1,flag,0] and gates ----
  if (tid < NTOK) {
    int t = tid;
    float p0, p1, flag;
    if (t < 64) {
      p0 = s[(size_t)b * 128 + t * 2 + 0];
      p1 = s[(size_t)b * 128 + t * 2 + 1];
      flag = 1.0f;
      float q0 = g[(size_t)b * 128 + t * 2 + 0];
      float q1 = g[(size_t)b * 128 + t * 2 + 1];
      sg[t] = ((p0 + p1) != 0.0f && (q0 + q1) != 0.0f) ? 1.0f : 0.0f;
    } else {
      int j = t - 64;
      p0 = g[(size_t)b * 128 + j * 2 + 0];
      p1 = g[(size_t)b * 128 + j * 2 + 1];
      flag = 0.0f;
    }
    buf0[t * LS + 0] = p0;
    buf0[t * LS + 1] = p1;
    buf0[t * LS + 2] = flag;
    buf0[t * LS + 3] = 0.0f;  // pad din 3 -> 4
  }
  __syncthreads();

  const int row0 = w * 16;
  const int m = lane & 15, h = lane >> 4, n = lane & 15;
  const int pw = w ^ 4;  // partner tile (token t <-> t^64)
  // A-fragment row pointers: element k at (aRow + k) holds X[row0+m][k + 2h-adjusted]
  const float* aR0 = buf0 + (row0 + m) * LS + 2 * h;
  const float* aS0 = buf1 + (row0 + m) * LS + 2 * h;
  const float* aP = buf1 + (pw * 16 + m) * LS + 2 * h;
  float* d0 = buf0 + row0 * LS;
  float* d1 = buf1 + row0 * LS;
  const float gate = sg[16 * (w & 3) + m];  // sg_pair[t % 64] for this lane's row

  v8f cR[4], cS[4];

  // ================= Stage 1 (gconv1): dims 4(3)->8->16->32, both chains everywhere ====
  // L1 (shared A from X)
  zeroAcc<1>(cR); zeroAcc<1>(cS);
  mm_dual<1, 1>(aR0, wsf + 0 + lane * 2, wsf + 768 + lane * 2, cR, cS);
  finish<1, true>(cR, wsf + 64 + 0, n);
  finish<1, true>(cS, wsf + 832 + 0, n);
  storeD<1>(d0, cR, lane);
  storeD<1>(d1, cS, lane);
  // L2
  zeroAcc<1>(cR); zeroAcc<1>(cS);
  mm_plain<2, 1>(aR0, wsf + 80 + lane * 2, cR);
  mm_plain<2, 1>(aS0, wsf + 848 + lane * 2, cS);
  finish<1, true>(cR, wsf + 208, n);
  finish<1, true>(cS, wsf + 976, n);
  storeD<1>(d0, cR, lane);
  storeD<1>(d1, cS, lane);
  // L3
  zeroAcc<2>(cR); zeroAcc<2>(cS);
  mm_plain<4, 2>(aR0, wsf + 224 + lane * 2, cR);
  mm_plain<4, 2>(aS0, wsf + 992 + lane * 2, cS);
  finish<2, true>(cR, wsf + 736, n);
  finish<2, true>(cS, wsf + 1504, n);
  storeD<2>(d0, cR, lane);
  storeD<2>(d1, cS, lane);
  __syncthreads();  // sup rows visible across waves for the adj swap

  // ================= Stage 2 (gconv2): dims 32->64->64->64 =================
  // Trim: x2 rows <64 need only root chain; rows >=64 only sup chain.
  const bool isRoot = (w < 4);
  const int oB1 = isRoot ? 1536 : 11968, ob1 = isRoot ? 3584 : 14016;
  const int oB2 = isRoot ? 3648 : 14080, ob2 = isRoot ? 7744 : 18176;
  const int oB3 = isRoot ? 7808 : 18240, ob3 = isRoot ? 11904 : 22336;
  const float* aMy = isRoot ? aR0 : aS0;
  float* dMy = isRoot ? d0 : d1;

  v8f c[4];
  // L1 on combined x1 = root1(own) + gate*sup1(partner)
  zeroAcc<4>(c);
  mm_comb<8, 4>(aR0, aP, gate, wsf + oB1 + lane * 2, c);
  __syncthreads();  // everyone done reading partner rows before overwrite
  finish<4, true>(c, wsf + ob1, n);
  storeD<4>(dMy, c, lane);
  // L2 (wave-private rows)
  zeroAcc<4>(c);
  mm_plain<16, 4>(aMy, wsf + oB2 + lane * 2, c);
  finish<4, true>(c, wsf + ob2, n);
  storeD<4>(dMy, c, lane);
  // L3
  zeroAcc<4>(c);
  mm_plain<16, 4>(aMy, wsf + oB3 + lane * 2, c);
  finish<4, true>(c, wsf + ob3, n);
  storeD<4>(dMy, c, lane);
  __syncthreads();  // sup2(partner) rows visible for the head's swap

  // ================= Head (rows < 64 only: waves 0..3) =================
  if (w < 4) {
    // fc1 on combined x2 = root2(own,buf0) + gate*sup2(partner,buf1), relu
    v8f hc[4];
    zeroAcc<4>(hc);
    mm_comb<16, 4>(aR0, aP, gate, wsf + 22400 + lane * 2, hc);
    finish<4, true>(hc, wsf + 26496, n);
    storeD<4>(d0, hc, lane);  // own rows, wave-private now
    // fc2 (dout 8 padded to 16), no relu
    v8f qa[1];
    zeroAcc<1>(qa);
    mm_plain<16, 1>(aR0, wsf + 26560 + lane * 2, qa);
    finish<1, false>(qa, wsf + 27584, n);
    if (n < 8) {
#pragma unroll
      for (int v = 0; v < 8; ++v) {
        int t = row0 + v + 8 * h;  // node 0..63
        out[((size_t)b * 64 + t) * 8 + n] = qa[0][v];
      }
    }
  }
}

// ---------------- launch ----------------
extern "C" void kernel_launch(void* const* d_in, const int* in_sizes, int n_in,
                              void* d_out, int out_size, void* d_ws, size_t ws_size,
                              hipStream_t stream) {
  (void)in_sizes; (void)n_in; (void)out_size; (void)ws_size;
  const float* s = (const float*)d_in[0];
  const float* g = (const float*)d_in[1];
  float* wsf = (float*)d_ws;  // needs 27600 floats (~110KB)

  // {W idx, b idx, din, dout, dinp, ntiles, Boff, biasOff}
  struct LayDesc { int wi, bi, din, dout, dinp, nt, bOff, biasOff; };
  const LayDesc L[14] = {
      {2, 3, 3, 8, 4, 1, 0, 64},        {4, 5, 8, 16, 8, 1, 80, 208},
      {6, 7, 16, 32, 16, 2, 224, 736},  {8, 9, 3, 8, 4, 1, 768, 832},
      {10, 11, 8, 16, 8, 1, 848, 976},  {12, 13, 16, 32, 16, 2, 992, 1504},
      {14, 15, 32, 64, 32, 4, 1536, 3584},  {16, 17, 64, 64, 64, 4, 3648, 7744},
      {18, 19, 64, 64, 64, 4, 7808, 11904}, {20, 21, 32, 64, 32, 4, 11968, 14016},
      {22, 23, 64, 64, 64, 4, 14080, 18176},{24, 25, 64, 64, 64, 4, 18240, 22336},
      {26, 27, 64, 64, 64, 4, 22400, 26496},{28, 29, 64, 8, 64, 1, 26560, 27584}};
  for (int i = 0; i < 14; ++i) {
    int Bsz = L[i].nt * (L[i].dinp / 4) * 64;
    int total = Bsz + L[i].nt * 16;
    prep_weights<<<(total + 255) / 256, 256, 0, stream>>>(
        (const float*)d_in[L[i].wi], (const float*)d_in[L[i].bi],
        wsf + L[i].bOff, wsf + L[i].biasOff, L[i].din, L[i].dout, L[i].dinp, L[i].nt);
  }

  const int smemBytes = (2 * NTOK * LS + 64) * (int)sizeof(float);  // 67,840 B (<320KB/WGP)
  hipFuncSetAttribute((const void*)gtq_fused,
                      hipFuncAttributeMaxDynamicSharedMemorySize, smemBytes);
  gtq_fused<<<8192, 256, smemBytes, stream>>>(s, g, wsf, (float*)d_out);
}